// VQC_14164802142364
// MI455X (gfx1250) — compile-verified
//
#include <hip/hip_runtime.h>

// ---------------------------------------------------------------------------
// VQC forward for MI455X (gfx1250, wave32).
// Kernel 1: one workgroup per batch element; 8192-amplitude complex state held
//           entirely in 64KB of LDS across all 3 layers; writes |psi|^2 to d_ws.
// Kernel 2: final projection probs(512x8192) @ Z(8192x16, cols 10..15 zero)
//           via V_WMMA_F32_16X16X4_F32, split-K over 4 waves per block.
// ---------------------------------------------------------------------------

#define N_WIRES 13
#define DIM     8192            // 2^13
#define NLAY    3
#define BATCH   512
#define NOUT    10
#define BLK     256
#define EPT     (DIM / BLK)       // 32 elements / thread
#define PPT     (DIM / 2 / BLK)   // 16 gate-pairs / thread
#define CPT     (DIM / 4 / BLK)   // 8 cnot-pairs / thread

typedef __attribute__((ext_vector_type(2))) float v2f;
typedef __attribute__((ext_vector_type(8))) float v8f;

__device__ __forceinline__ float zsign(int k, int pos) {
    // +1.0f if bit==0, -1.0f if bit==1
    unsigned bit = ((unsigned)k >> pos) & 1u;
    return __int_as_float(0x3f800000u | (bit << 31));
}

__global__ __launch_bounds__(BLK)
void vqc_state_kernel(const float* __restrict__ inputs,
                      const float* __restrict__ weights,
                      const float* __restrict__ embed,
                      float* __restrict__ probs)
{
    extern __shared__ float smem[];
    float* sre = smem;        // [DIM]
    float* sim = smem + DIM;  // [DIM]

    const int b   = blockIdx.x;
    const int tid = threadIdx.x;
    const float AMP0 = 0.011048543456039806f;  // 8192^-0.5

    for (int l = 0; l < NLAY; ++l) {
        // ---- per-layer embed coefficients into registers ----
        float cf[N_WIRES], gm[N_WIRES];
        #pragma unroll
        for (int w = 0; w < N_WIRES; ++w) {
            const float* e = embed + (size_t)(l * N_WIRES + w) * 3;
            cf[w] = inputs[b * N_WIRES + w] * e[0] + e[1];  // x*omega + bias
            gm[w] = e[2];
        }

        // ---- diagonal ZZ-embedding phase: psi *= exp(-i*ang) ----
        for (int t = 0; t < EPT; ++t) {
            const int k = tid + t * BLK;
            float ang = 0.f;
            const float z0 = zsign(k, 12);   // wire 0 is MSB
            float zp = z0;
            #pragma unroll
            for (int w = 0; w < N_WIRES - 1; ++w) {
                const float zn = zsign(k, 11 - w);   // wire w+1
                ang += cf[w] * zp + gm[w] * zp * zn;
                zp = zn;
            }
            ang += cf[N_WIRES - 1] * zp + gm[N_WIRES - 1] * zp * z0;
            float sn, cs;
            __sincosf(ang, &sn, &cs);
            float re, im;
            if (l == 0) { re = AMP0 * cs; im = -AMP0 * sn; }
            else {
                const float pr = sre[k], pi = sim[k];
                re = pr * cs + pi * sn;
                im = pi * cs - pr * sn;
            }
            sre[k] = re; sim[k] = im;
        }
        __syncthreads();

        // ---- 13 single-qubit Rot(phi,theta,omega) gates, in place ----
        for (int w = 0; w < N_WIRES; ++w) {
            const float* wp = weights + (size_t)(l * N_WIRES + w) * 3;
            const float phi = wp[0], th = wp[1], om = wp[2];
            float sh, ch, sa, ca, sb, cb;
            __sincosf(0.5f * th, &sh, &ch);
            __sincosf(0.5f * (phi + om), &sa, &ca);
            __sincosf(0.5f * (phi - om), &sb, &cb);
            const float u00r =  ch * ca, u00i = -ch * sa;
            const float u01r = -sh * cb, u01i = -sh * sb;
            const float u10r =  sh * cb, u10i = -sh * sb;
            const float u11r =  ch * ca, u11i =  ch * sa;
            const int p = 12 - w;
            const int stride = 1 << p;
            for (int t = 0; t < PPT; ++t) {
                const int q  = tid + t * BLK;               // 0..4095
                const int i0 = ((q & ~(stride - 1)) << 1) | (q & (stride - 1));
                const int i1 = i0 + stride;
                const float r0 = sre[i0], c0 = sim[i0];
                const float r1 = sre[i1], c1 = sim[i1];
                sre[i0] = u00r * r0 - u00i * c0 + u01r * r1 - u01i * c1;
                sim[i0] = u00r * c0 + u00i * r0 + u01r * c1 + u01i * r1;
                sre[i1] = u10r * r0 - u10i * c0 + u11r * r1 - u11i * c1;
                sim[i1] = u10r * c0 + u10i * r0 + u11r * c1 + u11i * r1;
            }
            __syncthreads();
        }

        // ---- ring of 13 CNOTs: swap amplitudes where control bit = 1 ----
        for (int w = 0; w < N_WIRES; ++w) {
            const int pc = 12 - w;
            const int pt = 12 - ((w + 1) % N_WIRES);
            const int hi = pc > pt ? pc : pt;
            const int lo = pc > pt ? pt : pc;
            for (int t = 0; t < CPT; ++t) {
                const int q    = tid + t * BLK;             // 11 free bits
                const int low  = q & ((1 << lo) - 1);
                const int mid  = (q >> lo) & ((1 << (hi - lo - 1)) - 1);
                const int high = q >> (hi - 1);
                int idx = low | (mid << (lo + 1)) | (high << (hi + 1));
                idx |= (1 << pc);                           // control set, target clear
                const int j = idx | (1 << pt);
                float tr = sre[idx]; sre[idx] = sre[j]; sre[j] = tr;
                float ti = sim[idx]; sim[idx] = sim[j]; sim[j] = ti;
            }
            __syncthreads();
        }
    }

    // ---- probabilities -> global (lives in 192MB L2 for kernel 2) ----
    for (int t = 0; t < EPT; ++t) {
        const int k = tid + t * BLK;
        const float re = sre[k], im = sim[k];
        probs[(size_t)b * DIM + k] = re * re + im * im;
    }
}

// ---------------------------------------------------------------------------
// out[b, o] = sum_k probs[b,k] * Z_TAB[k,o],  Z_TAB[k,o] = 1 - 2*((k>>(12-o))&1)
// M=512 (16/block), N=16 (10 live), K=8192 split over 4 waves per block.
// ---------------------------------------------------------------------------
__global__ __launch_bounds__(128)
void vqc_project_kernel(const float* __restrict__ probs, float* __restrict__ out)
{
    __shared__ float red[4][8][32];

    const int lane  = threadIdx.x & 31;
    const int wv    = threadIdx.x >> 5;
    const int m0    = blockIdx.x * 16;
    const int row   = m0 + (lane & 15);
    const int khalf = (lane >> 4) << 1;          // K offset 0 (lanes 0-15) or 2 (16-31)
    const int n     = lane & 15;                 // output column for B/D
    const int sh    = (n < NOUT) ? (12 - n) : 0; // sign bit position (>=3)
    const bool live = (n < NOUT);

    const float* prow = probs + (size_t)row * DIM;
    v8f acc = {};

    const int kbeg = wv * (DIM / 4);
    const int kend = kbeg + (DIM / 4);
    for (int k = kbeg; k < kend; k += 4) {
        // A tile 16x4 f32: lane L -> M = L&15; VGPR0/1 -> K = khalf, khalf+1
        const int kb = k + khalf;
        v2f a = *reinterpret_cast<const v2f*>(prow + kb);  // 8B aligned
        // B tile 4x16: sign depends only on bits >= 3 of k for n<10, so it is
        // identical across the 4 K-rows of this tile.
        float z = live ? zsign(k, sh) : 0.f;
        v2f bv; bv.x = z; bv.y = z;
        acc = __builtin_amdgcn_wmma_f32_16x16x4_f32(
            /*neg_a=*/false, a, /*neg_b=*/false, bv,
            /*c_mod=*/(short)0, acc, /*reuse_a=*/false, /*reuse_b=*/false);
    }

    // split-K reduction across the 4 waves
    #pragma unroll
    for (int i = 0; i < 8; ++i) red[wv][i][lane] = acc[i];
    __syncthreads();

    if (wv == 0) {
        #pragma unroll
        for (int i = 0; i < 8; ++i) {
            const float s = red[0][i][lane] + red[1][i][lane]
                          + red[2][i][lane] + red[3][i][lane];
            const int rrow = i + ((lane >> 4) << 3);   // lanes 16-31 hold M = i+8
            const int col  = lane & 15;
            if (col < NOUT)
                out[(size_t)(m0 + rrow) * NOUT + col] = s;
        }
    }
}

extern "C" void kernel_launch(void* const* d_in, const int* in_sizes, int n_in,
                              void* d_out, int out_size, void* d_ws, size_t ws_size,
                              hipStream_t stream) {
    const float* inputs  = (const float*)d_in[0];  // (512, 13)
    const float* weights = (const float*)d_in[1];  // (3, 13, 3)
    const float* embed   = (const float*)d_in[2];  // (3, 13, 3)
    float* out   = (float*)d_out;                  // (512, 10)
    float* probs = (float*)d_ws;                   // 512*8192*4 = 16 MB scratch

    const size_t lds_bytes = 2 * DIM * sizeof(float);  // 64 KB dynamic LDS
    vqc_state_kernel<<<BATCH, BLK, lds_bytes, stream>>>(inputs, weights, embed, probs);
    vqc_project_kernel<<<BATCH / 16, 128, 0, stream>>>(probs, out);
}